// GCN_18202071400723
// MI455X (gfx1250) — compile-verified
//
#include <hip/hip_runtime.h>

typedef __attribute__((ext_vector_type(2))) float v2f;
typedef __attribute__((ext_vector_type(8))) float v8f;

#define IN_DIM 128
#define HID_DIM 64

// ---------------- utility kernels ----------------

__global__ void gcn_zero_kernel(float* __restrict__ p, long long n) {
    long long i = (long long)blockIdx.x * blockDim.x + threadIdx.x;
    long long stride = (long long)gridDim.x * blockDim.x;
    for (; i < n; i += stride) p[i] = 0.0f;
}

__global__ void gcn_degree_kernel(const int* __restrict__ dst,
                                  float* __restrict__ deg, int E) {
    int e = blockIdx.x * blockDim.x + threadIdx.x;
    if (e < E) atomicAdd(&deg[dst[e]], 1.0f);
}

__global__ void gcn_norm_kernel(float* __restrict__ deg_norm, int N) {
    int i = blockIdx.x * blockDim.x + threadIdx.x;
    if (i < N) {
        float d = deg_norm[i];
        deg_norm[i] = rsqrtf(fmaxf(d, 1.0f));   // clamp(deg,1)^-0.5
    }
}

__global__ void gcn_relu_kernel(float* __restrict__ p, long long n) {
    long long i = (long long)blockIdx.x * blockDim.x + threadIdx.x;
    long long stride = (long long)gridDim.x * blockDim.x;
    for (; i < n; i += stride) p[i] = fmaxf(p[i], 0.0f);
}

// ---------------- WMMA f32 GEMM: H[M,64] = act(X[M,K]) @ W[K,64] ----------------
// One wave computes a 16x16 tile of H via V_WMMA_F32_16X16X4_F32.
// Block = 128 threads (4 waves) -> covers all 64 output columns.
// Grid  = M/16 blocks (M = 50000 = 3125*16, no tail).
//
// ISA layouts (wave32):
//   A 16x4 f32 : lanes 0-15 M=0..15 ; VGPR0 = K{0 | 2}, VGPR1 = K{1 | 3} (lane half selects)
//   B 4x16 f32 : VGPRr lanes 0-15 = row K=r, lanes 16-31 = row K=r+2, N = lane&15
//   C 16x16 f32: VGPRr lanes 0-15 = M=r, lanes 16-31 = M=r+8, N = lane&15
template <int K, bool RELU_A>
__global__ void gcn_gemm_wmma(const float* __restrict__ X,
                              const float* __restrict__ W,
                              float* __restrict__ H) {
    __shared__ float sW[K * HID_DIM];
    for (int i = threadIdx.x; i < K * HID_DIM; i += blockDim.x) sW[i] = W[i];
    __syncthreads();

    const int wave = threadIdx.x >> 5;
    const int lane = threadIdx.x & 31;
    const int half = lane >> 4;        // 0: lanes 0-15, 1: lanes 16-31
    const int lid  = lane & 15;
    const int row0 = blockIdx.x * 16;
    const int col0 = wave * 16;

    const float* __restrict__ Arow = X + (size_t)(row0 + lid) * K;

    v8f c = {};
#pragma unroll
    for (int k0 = 0; k0 < K; k0 += 4) {
        v2f a = *(const v2f*)(Arow + k0 + 2 * half);   // K = k0+2h, k0+2h+1
        if (RELU_A) {
            a.x = fmaxf(a.x, 0.0f);
            a.y = fmaxf(a.y, 0.0f);
        }
        v2f b;
        b.x = sW[(k0 + 2 * half) * HID_DIM + col0 + lid];
        b.y = sW[(k0 + 2 * half + 1) * HID_DIM + col0 + lid];
        c = __builtin_amdgcn_wmma_f32_16x16x4_f32(
                /*neg_a=*/false, a, /*neg_b=*/false, b,
                /*c_mod=*/(short)0, c, /*reuse_a=*/false, /*reuse_b=*/false);
    }

    float* __restrict__ Hb = H + (size_t)row0 * HID_DIM + col0;
#pragma unroll
    for (int r = 0; r < 8; ++r)
        Hb[(size_t)(r + 8 * half) * HID_DIM + lid] = c[r];
}

// ---------------- edge scatter: agg[dst] += H[src] * norm[src] ----------------
// 16 lanes per edge; each lane moves a float4 (coalesced 256B/edge) and issues
// 4 global_atomic_add_f32.
__global__ void gcn_scatter_kernel(const float4* __restrict__ H4,
                                   const float* __restrict__ norm,
                                   const int* __restrict__ src,
                                   const int* __restrict__ dst,
                                   float* __restrict__ agg, int E) {
    long long gid = (long long)blockIdx.x * blockDim.x + threadIdx.x;
    int e = (int)(gid >> 4);
    int q = (int)(gid & 15);
    if (e >= E) return;
    int s = src[e];
    int t = dst[e];
    float4 v = H4[(size_t)s * (HID_DIM / 4) + q];
    float n = norm[s];
    float* o = agg + (size_t)t * HID_DIM + q * 4;
    atomicAdd(o + 0, v.x * n);
    atomicAdd(o + 1, v.y * n);
    atomicAdd(o + 2, v.z * n);
    atomicAdd(o + 3, v.w * n);
}

// ---------------- launcher ----------------

extern "C" void kernel_launch(void* const* d_in, const int* in_sizes, int n_in,
                              void* d_out, int out_size, void* d_ws, size_t ws_size,
                              hipStream_t stream) {
    const float* features = (const float*)d_in[0];   // [N, 128]
    const float* W1       = (const float*)d_in[1];   // [128, 64]
    const float* W2       = (const float*)d_in[2];   // [64, 64]
    const int*   src      = (const int*)d_in[3];     // [E]
    const int*   dst      = (const int*)d_in[4];     // [E]

    const int N = in_sizes[0] / IN_DIM;              // 50000
    const int E = in_sizes[3];                       // 800000
    const long long NH = (long long)N * HID_DIM;     // 3.2M floats

    float* ws   = (float*)d_ws;
    float* norm = ws;                                // [N] degree -> norm
    float* H    = ws + 65536;                        // [N, 64] GEMM output
    float* AGG  = H + NH;                            // [N, 64] layer-1 aggregate
    float* OUT  = (float*)d_out;                     // [N, 64]

    const int mtiles = N / 16;                       // 3125

    // degree -> norm
    gcn_zero_kernel<<<512, 256, 0, stream>>>(norm, N);
    gcn_degree_kernel<<<(E + 255) / 256, 256, 0, stream>>>(dst, norm, E);
    gcn_norm_kernel<<<(N + 255) / 256, 256, 0, stream>>>(norm, N);

    // ---- layer 1 ----
    gcn_gemm_wmma<IN_DIM, false><<<mtiles, 128, 0, stream>>>(features, W1, H);
    gcn_zero_kernel<<<2048, 256, 0, stream>>>(AGG, NH);
    gcn_scatter_kernel<<<(E * 16 + 255) / 256, 256, 0, stream>>>(
        (const float4*)H, norm, src, dst, AGG, E);
    // relu(agg) fused into layer-2 GEMM A-load (relu(relu(x)) == relu(x))

    // ---- layer 2 ----
    gcn_gemm_wmma<HID_DIM, true><<<mtiles, 128, 0, stream>>>(AGG, W2, H);
    gcn_zero_kernel<<<2048, 256, 0, stream>>>(OUT, NH);
    gcn_scatter_kernel<<<(E * 16 + 255) / 256, 256, 0, stream>>>(
        (const float4*)H, norm, src, dst, OUT, E);
    gcn_relu_kernel<<<2048, 256, 0, stream>>>(OUT, NH);
}